// LDRMultiLayerModel_10419590660373
// MI455X (gfx1250) — compile-verified
//
#include <hip/hip_runtime.h>
#include <hip/hip_bf16.h>
#include <math.h>

// ---------------- problem constants ----------------
#define BATCH   128
#define NCH     3
#define NPIX    32
#define IMG     (NPIX*NPIX)        // 1024
#define NFILT   16                 // J*L morlet filters
#define NPATH   81
#define FEAT    15552              // 3*81*8*8
#define RANK    16
#define FFTN    32768
#define NUMCLS  10
#define TWO_PI  6.28318530717958647692f

typedef unsigned int u32;
typedef __attribute__((ext_vector_type(2))) float v2f;
typedef __attribute__((ext_vector_type(8))) float v8f;
typedef __attribute__((ext_vector_type(4))) u32  v4u;
typedef __attribute__((ext_vector_type(4))) int  v4i;
typedef __attribute__((ext_vector_type(8))) int  v8i;

static __device__ __forceinline__ float2 cmul(float2 a, float2 b) {
  return make_float2(a.x*b.x - a.y*b.y, a.x*b.y + a.y*b.x);
}
static __device__ __forceinline__ float2 cmulj(float2 a, float2 b) { // a*conj(b)
  return make_float2(a.x*b.x + a.y*b.y, a.y*b.x - a.x*b.y);
}
static __device__ __forceinline__ float2 cadd(float2 a, float2 b) {
  return make_float2(a.x + b.x, a.y + b.y);
}

// ============================================================
// Tensor Data Mover: contiguous 256KB tile  global <-> LDS.
// D# per CDNA5 ISA ch.8: group0 = {count=1 | lds_addr | global_addr | type=2},
// group1 = {data_size=8B, tensor/tile dim0 = 32768, dim1 = 1}.
// Tracked on TENSORcnt; issued once per workgroup (scalar-uniform branch).
// ============================================================
#if __has_builtin(__builtin_amdgcn_tensor_load_to_lds)
#define HAVE_TDM 1
#else
#define HAVE_TDM 0
#endif

#if HAVE_TDM
static __device__ __forceinline__ void tdm_desc(const void* gaddr, u32 lds_off,
                                                v4u* g0, v8i* g1)
{
  unsigned long long ga = (unsigned long long)(uintptr_t)gaddr;
  *g0 = (v4u){ 1u,                                   // count=1 valid user D#
               lds_off,                               // lds_addr (bytes)
               (u32)(ga & 0xFFFFFFFFu),               // global_addr[31:0]
               ((u32)(ga >> 32) & 0x01FFFFFFu) | (2u << 30) };  // addr[56:32] | type=2
  *g1 = (v8i){ (int)(3u << 16),                      // data_size = 3 (8 bytes)
               (int)0x80000000u,                      // tensor_dim0 = 32768 (bits 63:48)
               (int)0x00010000u,                      // tensor_dim1 = 1    (bits 95:80)
               (int)0x80000000u,                      // tile_dim0  = 32768 (bits 127:112)
               1,                                     // tile_dim1 = 1      (bits 143:128)
               32768,                                 // tensor_dim0_stride lo32
               (int)0x80000000u,                      // tensor_dim1_stride lo16 (bits 223:208)
               0 };
}
static __device__ __forceinline__ void tdm_load_g2l(const void* gsrc, u32 lds_off)
{
  v4u g0; v8i g1; tdm_desc(gsrc, lds_off, &g0, &g1);
  v4i z = {0,0,0,0};
#if __clang_major__ >= 23
  v8i z8 = {0,0,0,0,0,0,0,0};
  __builtin_amdgcn_tensor_load_to_lds(g0, g1, z, z, z8, 0);
#else
  __builtin_amdgcn_tensor_load_to_lds(g0, g1, z, z, 0);
#endif
}
static __device__ __forceinline__ void tdm_store_l2g(void* gdst, u32 lds_off)
{
  v4u g0; v8i g1; tdm_desc(gdst, lds_off, &g0, &g1);
  v4i z = {0,0,0,0};
#if __clang_major__ >= 23
  v8i z8 = {0,0,0,0,0,0,0,0};
  __builtin_amdgcn_tensor_store_from_lds(g0, g1, z, z, z8, 0);
#else
  __builtin_amdgcn_tensor_store_from_lds(g0, g1, z, z, 0);
#endif
}
#endif

static __device__ __forceinline__ bool is_wave0() {
  return __builtin_amdgcn_readfirstlane(threadIdx.x >> 5) == 0;  // scalar branch
}

// ============================================================
// 32768-point complex FFT, LDS resident, mixed radix 16*16*16*8.
// Radix-16 butterflies = batched DFT-16 matmuls on V_WMMA_F32_16X16X4_F32.
// DIF forward: natural in -> digit-reversed out (post-twiddle).
// DIT inverse: digit-reversed in -> natural out  (pre-twiddle, conjugate).
// All k-space arrays live in the same digit-reversed order, so the
// elementwise spectral products need no reordering, and every
// global<->LDS spectrum transfer is a contiguous block copy (TDM-able).
// ============================================================
static __device__ __forceinline__ void build_w16(float2* w16, float dsign)
{
  if (threadIdx.x < 256) {
    int t = threadIdx.x & 15, q = threadIdx.x >> 4;
    float sn, cs;
    __sincosf(dsign * TWO_PI * (float)(t * q) * (1.0f / 16.0f), &sn, &cs);
    w16[threadIdx.x] = make_float2(cs, sn);
  }
  __syncthreads();
}

template <bool PRETW, int LGS>
static __device__ __forceinline__ void radix16_stage(float2* __restrict__ s,
                                                     const float2* __restrict__ w16,
                                                     float dsign)
{
  const int tid  = threadIdx.x;     // 512 = 16 wave32 waves
  const int lane = tid & 31;
  const int wave = tid >> 5;
  const int n16  = lane & 15;
  const int hi   = lane >> 4;
  const float wN = dsign * TWO_PI / (float)(16 << LGS);
  for (int chunk = wave; chunk < 128; chunk += 16) {  // uniform per wave: EXEC all 1s
    const int g = (chunk << 4) + n16;                 // this lane's column (group id)
    const int jj = g & ((1 << LGS) - 1);
    const int base = ((g >> LGS) << (LGS + 4)) + jj;
    v8f Dr = {0.f,0.f,0.f,0.f,0.f,0.f,0.f,0.f};
    v8f Di = {0.f,0.f,0.f,0.f,0.f,0.f,0.f,0.f};
#pragma unroll
    for (int kc = 0; kc < 4; ++kc) {
      const int ka = (kc << 2) + (hi << 1);           // K pair this lane half supplies
      float2 w0 = w16[n16 * 16 + ka];                 // A[m=n16][ka], A[m][ka+1]
      float2 w1 = w16[n16 * 16 + ka + 1];
      v2f Ar  = {  w0.x,  w1.x };
      v2f Ai  = {  w0.y,  w1.y };
      v2f AiN = { -w0.y, -w1.y };                     // f32 WMMA has no A-negate modifier
      float2 b0 = s[base + (ka << LGS)];
      float2 b1 = s[base + ((ka + 1) << LGS)];
      if (PRETW) {                                    // DIT: twiddle inputs
        float sn0, cs0, sn1, cs1;
        __sincosf(wN * (float)(jj * ka), &sn0, &cs0);
        __sincosf(wN * (float)(jj * (ka + 1)), &sn1, &cs1);
        b0 = make_float2(b0.x*cs0 - b0.y*sn0, b0.x*sn0 + b0.y*cs0);
        b1 = make_float2(b1.x*cs1 - b1.y*sn1, b1.x*sn1 + b1.y*cs1);
      }
      v2f Br = { b0.x, b1.x };
      v2f Bi = { b0.y, b1.y };
      // complex 16x16 DFT matmul: Dr += Ar*Br - Ai*Bi ; Di += Ar*Bi + Ai*Br
      Dr = __builtin_amdgcn_wmma_f32_16x16x4_f32(false, Ar,  false, Br, (short)0, Dr, false, false);
      Dr = __builtin_amdgcn_wmma_f32_16x16x4_f32(false, AiN, false, Bi, (short)0, Dr, false, false);
      Di = __builtin_amdgcn_wmma_f32_16x16x4_f32(false, Ar,  false, Bi, (short)0, Di, false, false);
      Di = __builtin_amdgcn_wmma_f32_16x16x4_f32(false, Ai,  false, Br, (short)0, Di, false, false);
    }
#pragma unroll
    for (int vr = 0; vr < 8; ++vr) {
      const int q = vr + (hi << 3);
      float dr = Dr[vr], di = Di[vr];
      if (!PRETW) {                                   // DIF: twiddle outputs
        float sn, cs;
        __sincosf(wN * (float)(jj * q), &sn, &cs);
        float tr = dr * cs - di * sn, ti = dr * sn + di * cs;
        dr = tr; di = ti;
      }
      s[base + (q << LGS)] = make_float2(dr, di);
    }
  }
  __syncthreads();
}

static __device__ __forceinline__ void radix8_stage(float2* __restrict__ s,
                                                    const float2* __restrict__ w16)
{
  for (int g = threadIdx.x; g < 4096; g += 512) {
    const int base = g << 3;
    float2 v[8], r[8];
#pragma unroll
    for (int t = 0; t < 8; ++t) v[t] = s[base + t];
#pragma unroll
    for (int q = 0; q < 8; ++q) {
      float2 acc = v[0];
#pragma unroll
      for (int t = 1; t < 8; ++t)        // w8^{qt} = w16^{2qt mod 16}
        acc = cadd(acc, cmul(v[t], w16[16 + ((2 * q * t) & 15)]));
      r[q] = acc;
    }
#pragma unroll
    for (int q = 0; q < 8; ++q) s[base + q] = r[q];
  }
  __syncthreads();
}

// forward: natural order in, digit-reversed out
static __device__ void fft_dif(float2* __restrict__ s, float2* __restrict__ w16)
{
  build_w16(w16, -1.0f);
  radix16_stage<false, 11>(s, w16, -1.0f);
  radix16_stage<false, 7 >(s, w16, -1.0f);
  radix16_stage<false, 3 >(s, w16, -1.0f);
  radix8_stage(s, w16);
}

// inverse (un-normalized F^H): digit-reversed in, natural order out
static __device__ void fft_dit(float2* __restrict__ s, float2* __restrict__ w16)
{
  build_w16(w16, +1.0f);
  radix8_stage(s, w16);
  radix16_stage<true, 3 >(s, w16, +1.0f);
  radix16_stage<true, 7 >(s, w16, +1.0f);
  radix16_stage<true, 11>(s, w16, +1.0f);
}

#define FFT_SMEM ((FFTN + 256) * sizeof(float2))   // 264 KB of the 320 KB WGP LDS

// ---------------- FFT wrapper kernels ----------------
// forward spectrum of packed row, in place; TDM block copy both ways
__global__ void __launch_bounds__(512) k_fft_fwd_inplace(float2* __restrict__ data)
{
  extern __shared__ float2 sm[];
  float2* s = sm; float2* w16 = sm + FFTN;
  float2* row = data + (size_t)blockIdx.x * FFTN;
#if HAVE_TDM
  if (is_wave0()) {
    tdm_load_g2l(row, (u32)(uintptr_t)s);
    __builtin_amdgcn_s_wait_tensorcnt(0);
  }
  // barrier inside build_w16 orders the DMA'd data for all waves
#else
  for (int k = threadIdx.x; k < FFTN; k += blockDim.x) {
    __builtin_prefetch(row + k + 512, 0, 1);
    s[k] = row[k];
  }
#endif
  fft_dif(s, w16);   // ends with __syncthreads()
#if HAVE_TDM
  if (is_wave0()) {
    tdm_store_l2g(row, (u32)(uintptr_t)s);
    __builtin_amdgcn_s_wait_tensorcnt(0);
  }
#else
  for (int k = threadIdx.x; k < FFTN; k += blockDim.x) row[k] = s[k];
#endif
}

// c(b,i) = ifft( Af(b) * conj(Hf(i)) ), keep real part truncated to FEAT
__global__ void __launch_bounds__(512) k_ifft_mulconj(const float2* __restrict__ Af,
                                                      const float2* __restrict__ Hf,
                                                      float* __restrict__ creal)
{
  extern __shared__ float2 sm[];
  float2* s = sm; float2* w16 = sm + FFTN;
  const int b = blockIdx.x >> 4, i = blockIdx.x & 15;
  const float2* ar = Af + (size_t)b * FFTN;
  const float2* hr = Hf + (size_t)i * FFTN;
  for (int k = threadIdx.x; k < FFTN; k += blockDim.x) {
    __builtin_prefetch(ar + k + 512, 0, 1);
    s[k] = cmulj(ar[k], hr[k]);        // both operands in matching (permuted) order
  }
  fft_dit(s, w16);
  float* cr = creal + (size_t)blockIdx.x * FEAT;
  const float inv = 1.0f / (float)FFTN;
  for (int j = threadIdx.x; j < FEAT; j += blockDim.x)
    cr[j] = s[j].x * inv;              // natural order out of DIT
}

// acc(b) = sum_i fft(zpad(c(b,i))) * Gf(i)  -- one block per b: deterministic
__global__ void __launch_bounds__(512) k_fft_mulacc(const float* __restrict__ creal,
                                                    const float2* __restrict__ Gf,
                                                    float2* __restrict__ acc)
{
  extern __shared__ float2 sm[];
  float2* s = sm; float2* w16 = sm + FFTN;
  const int b = blockIdx.x;
  float2* ab = acc + (size_t)b * FFTN;
  for (int i = 0; i < RANK; ++i) {
    __syncthreads();   // previous iteration's readers of s[] are done
    const float* cr = creal + ((size_t)b * RANK + i) * FEAT;
    for (int k = threadIdx.x; k < FFTN; k += blockDim.x) {
      __builtin_prefetch(cr + k + 512, 0, 1);
      s[k] = make_float2(k < FEAT ? cr[k] : 0.0f, 0.0f);
    }
    fft_dif(s, w16);
    const float2* gr = Gf + (size_t)i * FFTN;
    for (int k = threadIdx.x; k < FFTN; k += blockDim.x) {
      float2 v = cmul(s[k], gr[k]);    // matching permuted order
      if (i == 0) ab[k] = v;
      else {
        float2 o = ab[k];
        ab[k] = make_float2(o.x + v.x, o.y + v.y);
      }
    }
  }
}

// y(b) = real(ifft(acc(b)))[:FEAT] * P  (+optional ReLU)
__global__ void __launch_bounds__(512) k_ifft_final(const float2* __restrict__ acc,
                                                    const float* __restrict__ P,
                                                    float* __restrict__ fdst, int relu)
{
  extern __shared__ float2 sm[];
  float2* s = sm; float2* w16 = sm + FFTN;
  const float2* ar = acc + (size_t)blockIdx.x * FFTN;
#if HAVE_TDM
  if (is_wave0()) {
    tdm_load_g2l(ar, (u32)(uintptr_t)s);
    __builtin_amdgcn_s_wait_tensorcnt(0);
  }
#else
  for (int k = threadIdx.x; k < FFTN; k += blockDim.x) s[k] = ar[k];
#endif
  fft_dit(s, w16);
  float* fb = fdst + (size_t)blockIdx.x * FEAT;
  const float inv = 1.0f / (float)FFTN;
  for (int j = threadIdx.x; j < FEAT; j += blockDim.x) {
    float v = s[j].x * inv * P[j];
    fb[j] = relu ? fmaxf(v, 0.0f) : v;
  }
}

// ============================================================
// Scattering transform (32x32 grid, direct 32-pt DFT passes)
// ============================================================
static __device__ void dft2_32(float2* __restrict__ a, float2* __restrict__ t,
                               const float2* __restrict__ w32, int inv, bool scale)
{
  const int tid = threadIdx.x;           // 1024 threads: one element each
  const int row = tid >> 5, col = tid & 31;
  __syncthreads();                       // make prior elementwise writes to a[] visible
  float2 acc = make_float2(0.f, 0.f);
  for (int bb = 0; bb < 32; ++bb) {
    float2 w = w32[(col * bb) & 31];
    if (inv) w.y = -w.y;
    acc = cadd(acc, cmul(a[(row << 5) + bb], w));
  }
  t[tid] = acc;
  __syncthreads();
  acc = make_float2(0.f, 0.f);
  for (int r = 0; r < 32; ++r) {
    float2 w = w32[(row * r) & 31];
    if (inv) w.y = -w.y;
    acc = cadd(acc, cmul(t[(r << 5) + col], w));
  }
  if (scale) { acc.x *= (1.0f / 1024.0f); acc.y *= (1.0f / 1024.0f); }
  a[tid] = acc;                          // own element; next dft2 entry sync protects readers
}

// Build Morlet bank psi_hat (16x1024) and phi_hat (1024) by direct 2D DFT
__global__ void k_filters(float2* __restrict__ psi_hat, float2* __restrict__ phi_hat)
{
  __shared__ float r0[256], r1[256], r2[256];
  __shared__ float2 Kc; __shared__ float snorm;
  const int tid = threadIdx.x;
  const int fidx = blockIdx.x;           // 0..15 psi, 16 phi
  const bool isphi = (fidx == 16);
  float sigma, xi = 0.f, ct = 1.f, st = 0.f;
  if (isphi) { sigma = 0.8f * 4.0f; }
  else {
    int j = fidx >> 3, l = fidx & 7;
    sigma = 0.8f * (float)(1 << j);
    xi = (3.0f * (float)M_PI / 4.0f) / (float)(1 << j);
    float th = (float)l * (float)M_PI / 8.0f;
    ct = cosf(th); st = sinf(th);
  }
  const float is2 = 1.0f / (2.0f * sigma * sigma);
  float gx = 0.f, gy = 0.f, es = 0.f;
  for (int p = tid; p < IMG; p += 256) {
    int a = p >> 5, bb = p & 31;
    float y = (a  > 16) ? (float)(a  - 32) : (float)a;
    float x = (bb > 16) ? (float)(bb - 32) : (float)bb;
    float env = __expf(-(x * x + y * y) * is2);
    if (!isphi) {
      float sn, cs; __sincosf(xi * (x * ct + y * st), &sn, &cs);
      gx += env * cs; gy += env * sn;
    }
    es += env;
  }
  r0[tid] = gx; r1[tid] = gy; r2[tid] = es;
  __syncthreads();
  for (int off = 128; off > 0; off >>= 1) {
    if (tid < off) { r0[tid] += r0[tid+off]; r1[tid] += r1[tid+off]; r2[tid] += r2[tid+off]; }
    __syncthreads();
  }
  if (tid == 0) {
    float e = r2[0];
    Kc = make_float2(r0[0] / e, r1[0] / e);
    snorm = 1.0f / (TWO_PI * sigma * sigma);
  }
  __syncthreads();
  const float2 K = Kc; const float nrm = snorm;
  float2* dst = isphi ? phi_hat : (psi_hat + (size_t)fidx * IMG);
  for (int q = tid; q < IMG; q += 256) {
    int k1 = q >> 5, k2 = q & 31;
    float2 av = make_float2(0.f, 0.f);
    for (int p = 0; p < IMG; ++p) {
      int a = p >> 5, bb = p & 31;
      float y = (a  > 16) ? (float)(a  - 32) : (float)a;
      float x = (bb > 16) ? (float)(bb - 32) : (float)bb;
      float env = __expf(-(x * x + y * y) * is2);
      float pr, pi;
      if (isphi) { pr = env * nrm; pi = 0.f; }
      else {
        float sn, cs; __sincosf(xi * (x * ct + y * st), &sn, &cs);
        pr = (env * cs - K.x * env) * nrm;
        pi = (env * sn - K.y * env) * nrm;
      }
      int idx = (k1 * a + k2 * bb) & 31;
      float sn2, cs2; __sincosf(-TWO_PI * (float)idx * (1.0f / 32.0f), &sn2, &cs2);
      av.x += pr * cs2 - pi * sn2;
      av.y += pr * sn2 + pi * cs2;
    }
    dst[q] = av;
  }
}

// orders 0 & 1; stores u1_hat for j1<8 (needed by order 2)
__global__ void __launch_bounds__(1024) k_scatter1(const float* __restrict__ x,
                                                   const float2* __restrict__ psi,
                                                   const float2* __restrict__ phi,
                                                   float* __restrict__ f,
                                                   float2* __restrict__ u1h_g)
{
  __shared__ float2 xh[IMG], wk[IMG], tp[IMG];
  __shared__ float2 w32[NPIX];
  const int tid = threadIdx.x;
  const int bc = blockIdx.x;             // b*3 + c
  if (tid < NPIX) { float sn, cs; __sincosf(-TWO_PI * (float)tid / 32.0f, &sn, &cs); w32[tid] = make_float2(cs, sn); }
  wk[tid] = make_float2(x[(size_t)bc * IMG + tid], 0.f);
  dft2_32(wk, tp, w32, 0, false);        // xh = fft2(x)
  xh[tid] = wk[tid];
  const int row = tid >> 5, col = tid & 31;
  float* fb = f + (size_t)(bc / 3) * FEAT + (size_t)(bc % 3) * (NPATH * 64);
  // order 0
  wk[tid] = cmul(xh[tid], phi[tid]);
  dft2_32(wk, tp, w32, 1, true);
  if (((row | col) & 3) == 0) fb[(row >> 2) * 8 + (col >> 2)] = wk[tid].x;
  // order 1
  for (int k = 0; k < NFILT; ++k) {
    wk[tid] = cmul(xh[tid], psi[(size_t)k * IMG + tid]);
    dft2_32(wk, tp, w32, 1, true);                         // u1 = ifft2(xh*psi)
    float m = sqrtf(wk[tid].x * wk[tid].x + wk[tid].y * wk[tid].y);
    wk[tid] = make_float2(m, 0.f);
    dft2_32(wk, tp, w32, 0, false);                        // u1_hat
    if (k < 8) u1h_g[((size_t)bc * 8 + k) * IMG + tid] = wk[tid];
    wk[tid] = cmul(wk[tid], phi[tid]);
    dft2_32(wk, tp, w32, 1, true);                         // low-pass
    if (((row | col) & 3) == 0) fb[(1 + k) * 64 + (row >> 2) * 8 + (col >> 2)] = wk[tid].x;
  }
}

// order 2: j1 in [0,8), j2 in [0,8) through the j=1 wavelets
__global__ void __launch_bounds__(1024) k_scatter2(const float2* __restrict__ u1h_g,
                                                   const float2* __restrict__ psi,
                                                   const float2* __restrict__ phi,
                                                   float* __restrict__ f)
{
  __shared__ float2 xh[IMG], wk[IMG], tp[IMG];
  __shared__ float2 w32[NPIX];
  const int tid = threadIdx.x;
  const int bc = blockIdx.x >> 3;
  const int j1 = blockIdx.x & 7;
  if (tid < NPIX) { float sn, cs; __sincosf(-TWO_PI * (float)tid / 32.0f, &sn, &cs); w32[tid] = make_float2(cs, sn); }
  xh[tid] = u1h_g[((size_t)bc * 8 + j1) * IMG + tid];
  const int row = tid >> 5, col = tid & 31;
  float* fb = f + (size_t)(bc / 3) * FEAT + (size_t)(bc % 3) * (NPATH * 64);
  for (int j2 = 0; j2 < 8; ++j2) {
    wk[tid] = cmul(xh[tid], psi[(size_t)(8 + j2) * IMG + tid]);
    dft2_32(wk, tp, w32, 1, true);
    float m = sqrtf(wk[tid].x * wk[tid].x + wk[tid].y * wk[tid].y);
    wk[tid] = make_float2(m, 0.f);
    dft2_32(wk, tp, w32, 0, false);
    wk[tid] = cmul(wk[tid], phi[tid]);
    dft2_32(wk, tp, w32, 1, true);
    if (((row | col) & 3) == 0)
      fb[(17 + j1 * 8 + j2) * 64 + (row >> 2) * 8 + (col >> 2)] = wk[tid].x;
  }
}

// ---------------- glue kernels ----------------
__global__ void k_bnrelu(float* __restrict__ f, const float* __restrict__ gamma,
                         const float* __restrict__ beta)
{
  int j = blockIdx.x * blockDim.x + threadIdx.x;
  if (j >= FEAT) return;
  float s = 0.f, s2 = 0.f;
  for (int b = 0; b < BATCH; ++b) { float v = f[(size_t)b * FEAT + j]; s += v; s2 += v * v; }
  float mu = s * (1.0f / BATCH);
  float var = s2 * (1.0f / BATCH) - mu * mu;
  float sc = gamma[j] * rsqrtf(var + 1e-5f);
  float sh = beta[j] - mu * sc;
  for (int b = 0; b < BATCH; ++b)
    f[(size_t)b * FEAT + j] = fmaxf(f[(size_t)b * FEAT + j] * sc + sh, 0.f);
}

__global__ void k_cumprod(const float* __restrict__ subd, float* __restrict__ P)
{
  if (threadIdx.x == 0 && blockIdx.x == 0) {
    float p = 1.0f; P[0] = 1.0f;
    for (int k = 1; k < FEAT; ++k) { p *= subd[k - 1]; P[k] = p; }
  }
}

__global__ void k_pack(const float* __restrict__ G, const float* __restrict__ H,
                       const float* __restrict__ f, const float* __restrict__ P,
                       float2* __restrict__ Gc, float2* __restrict__ Hc,
                       float2* __restrict__ Ac)
{
  const int r = blockIdx.x;              // 0..(2R+B-1)
  const float* src; float2* dst; int mul;
  if (r < RANK)          { src = G + (size_t)r * FEAT;              dst = Gc + (size_t)r * FFTN;              mul = 0; }
  else if (r < 2 * RANK) { src = H + (size_t)(r - RANK) * FEAT;     dst = Hc + (size_t)(r - RANK) * FFTN;     mul = 0; }
  else                   { src = f + (size_t)(r - 2 * RANK) * FEAT; dst = Ac + (size_t)(r - 2 * RANK) * FFTN; mul = 1; }
  for (int k = threadIdx.x; k < FFTN; k += blockDim.x) {
    float v = 0.f;
    if (k < FEAT) { float p = P[k]; v = mul ? src[k] * p : src[k] / p; }
    dst[k] = make_float2(v, 0.f);
  }
}

__global__ void k_out(const float* __restrict__ f, float* __restrict__ out)
{
  int i = blockIdx.x * blockDim.x + threadIdx.x;
  if (i < BATCH * NUMCLS) out[i] = f[(size_t)(i / NUMCLS) * FEAT + (i % NUMCLS)];
}

// ============================================================
// Orchestration. Workspace need ~= 245 MB.
// ============================================================
extern "C" void kernel_launch(void* const* d_in, const int* in_sizes, int n_in,
                              void* d_out, int out_size, void* d_ws, size_t ws_size,
                              hipStream_t stream)
{
  const float* x     = (const float*)d_in[0];
  const float* gamma = (const float*)d_in[1];
  const float* beta  = (const float*)d_in[2];
  const float* Gs    = (const float*)d_in[3];
  const float* Hs    = (const float*)d_in[4];
  const float* subds = (const float*)d_in[5];
  float* out = (float*)d_out;
  (void)in_sizes; (void)n_in; (void)out_size; (void)ws_size;

  char* ws = (char*)d_ws;
  size_t off = 0;
  auto alloc = [&](size_t bytes) -> char* {
    char* p = ws + off; off += (bytes + 255) & ~(size_t)255; return p;
  };
  float2* PSI = (float2*)alloc((size_t)NFILT * IMG * sizeof(float2));
  float2* PHI = (float2*)alloc((size_t)IMG * sizeof(float2));
  float2* U1H = (float2*)alloc((size_t)BATCH * NCH * 8 * IMG * sizeof(float2));
  float*  FA  = (float*) alloc((size_t)BATCH * FEAT * sizeof(float));
  float*  FB  = (float*) alloc((size_t)BATCH * FEAT * sizeof(float));
  float*  Pv  = (float*) alloc((size_t)FEAT * sizeof(float));
  float2* Gc  = (float2*)alloc((size_t)RANK * FFTN * sizeof(float2));
  float2* Hc  = (float2*)alloc((size_t)RANK * FFTN * sizeof(float2));
  float2* Ac  = (float2*)alloc((size_t)BATCH * FFTN * sizeof(float2));
  float2* ACC = (float2*)alloc((size_t)BATCH * FFTN * sizeof(float2));
  float*  CR  = (float*) alloc((size_t)BATCH * RANK * FEAT * sizeof(float));

  const size_t fsmem = FFT_SMEM;

  k_filters <<<dim3(17),            dim3(256),  0, stream>>>(PSI, PHI);
  k_scatter1<<<dim3(BATCH*NCH),     dim3(1024), 0, stream>>>(x, PSI, PHI, FA, U1H);
  k_scatter2<<<dim3(BATCH*NCH*8),   dim3(1024), 0, stream>>>(U1H, PSI, PHI, FA);
  k_bnrelu  <<<dim3((FEAT+255)/256),dim3(256),  0, stream>>>(FA, gamma, beta);

  float* cur = FA; float* nxt = FB;
  for (int l = 0; l < 3; ++l) {
    k_cumprod<<<dim3(1), dim3(1), 0, stream>>>(subds + (size_t)l * (FEAT - 1), Pv);
    k_pack<<<dim3(2 * RANK + BATCH), dim3(512), 0, stream>>>(
        Gs + (size_t)l * RANK * FEAT, Hs + (size_t)l * RANK * FEAT, cur, Pv, Gc, Hc, Ac);
    k_fft_fwd_inplace<<<dim3(RANK),  dim3(512), fsmem, stream>>>(Gc);
    k_fft_fwd_inplace<<<dim3(RANK),  dim3(512), fsmem, stream>>>(Hc);
    k_fft_fwd_inplace<<<dim3(BATCH), dim3(512), fsmem, stream>>>(Ac);
    k_ifft_mulconj<<<dim3(BATCH * RANK), dim3(512), fsmem, stream>>>(Ac, Hc, CR);
    k_fft_mulacc  <<<dim3(BATCH),        dim3(512), fsmem, stream>>>(CR, Gc, ACC);
    k_ifft_final  <<<dim3(BATCH),        dim3(512), fsmem, stream>>>(ACC, Pv, nxt, (l < 2) ? 1 : 0);
    float* t = cur; cur = nxt; nxt = t;
  }
  k_out<<<dim3((BATCH * NUMCLS + 255) / 256), dim3(256), 0, stream>>>(cur, out);
}